// NormalLoss_44478681317469
// MI455X (gfx1250) — compile-verified
//
#include <hip/hip_runtime.h>
#include <hip/hip_bf16.h>
#include <math.h>

// ---------------- problem constants (from reference setup_inputs) ----------
#define IMG_B 16
#define IMG_H 384
#define IMG_W 1280
#define HWC   (IMG_H * IMG_W)

// ---------------- tiling ---------------------------------------------------
#define TILE_W 64
#define TILE_H 16
#define HX (TILE_W + 2)      // 66 halo cols
#define HY (TILE_H + 2)      // 18 halo rows
#define LOFF 3               // halo col hc lives at LDS col hc+LOFF (16B align)
#define HSTR 72              // padded LDS row stride (floats); 72*4=288 % 16 == 0
#define PLANE (HY * HSTR)    // 1296 floats per plane
#define NPLANES 6            // pred x/y/z + gt x/y/z
#define NTHREADS 256         // 8 wave32s

// Per-row async transfer slots: [0]=left b32, [1..16]=16x b128, [17]=right b32
#define SLOTS_PER_ROW 18
#define SLOTS_PER_PLANE (HY * SLOTS_PER_ROW)          // 324
#define TOTAL_SLOTS (NPLANES * SLOTS_PER_PLANE)       // 1944

// Builtin signatures (hipcc-diagnostic-confirmed):
//   b32 : (global int*,  lds int*,  imm off, imm cpol)
//   b128: (global int4*, lds int4*, imm off, imm cpol)
typedef int v4i __attribute__((vector_size(16)));
typedef __attribute__((address_space(1))) int GASI;
typedef __attribute__((address_space(3))) int LASI;
typedef __attribute__((address_space(1))) v4i GASV4;
typedef __attribute__((address_space(3))) v4i LASV4;

__device__ __forceinline__ void async_copy_b32(const float* src, float* dst) {
#if __has_builtin(__builtin_amdgcn_global_load_async_to_lds_b32)
  __builtin_amdgcn_global_load_async_to_lds_b32((GASI*)src, (LASI*)dst, 0, 0);
#else
  *dst = *src;
#endif
}

__device__ __forceinline__ void async_copy_b128(const float* src, float* dst) {
#if __has_builtin(__builtin_amdgcn_global_load_async_to_lds_b128)
  __builtin_amdgcn_global_load_async_to_lds_b128((GASV4*)src, (LASV4*)dst, 0, 0);
#elif __has_builtin(__builtin_amdgcn_global_load_async_to_lds_b32)
  __builtin_amdgcn_global_load_async_to_lds_b32((GASI*)(src + 0), (LASI*)(dst + 0), 0, 0);
  __builtin_amdgcn_global_load_async_to_lds_b32((GASI*)(src + 1), (LASI*)(dst + 1), 0, 0);
  __builtin_amdgcn_global_load_async_to_lds_b32((GASI*)(src + 2), (LASI*)(dst + 2), 0, 0);
  __builtin_amdgcn_global_load_async_to_lds_b32((GASI*)(src + 3), (LASI*)(dst + 3), 0, 0);
#else
  *(float4*)dst = *(const float4*)src;
#endif
}

__device__ __forceinline__ void wait_async_zero() {
#if __has_builtin(__builtin_amdgcn_s_wait_asynccnt)
  __builtin_amdgcn_s_wait_asynccnt(0);
#elif __has_builtin(__builtin_amdgcn_global_load_async_to_lds_b32)
  asm volatile("s_wait_asynccnt 0" ::: "memory");
#endif
}

__global__ __launch_bounds__(NTHREADS)
void normal_loss_tiles(const float* __restrict__ pred,
                       const float* __restrict__ gt,
                       const float* __restrict__ mask,
                       float* __restrict__ partial, int nblocks) {
  __shared__ float sm[NPLANES * PLANE];   // ~30.4 KB
  __shared__ float red[2 * NTHREADS];

  const int tx0 = blockIdx.x * TILE_W;
  const int ty0 = blockIdx.y * TILE_H;
  const int b   = blockIdx.z;
  const int tid = threadIdx.x;

  // ---- border blocks: vector-zero the halo once (conv zero padding) -------
  const bool border = (blockIdx.x == 0) | (blockIdx.x == gridDim.x - 1) |
                      (blockIdx.y == 0) | (blockIdx.y == gridDim.y - 1);
  if (border) {
    float4* smv = (float4*)sm;
    for (int i = tid; i < NPLANES * PLANE / 4; i += NTHREADS)
      smv[i] = make_float4(0.f, 0.f, 0.f, 0.f);
    __syncthreads();
  }

  // ---- stage 6 halo planes into LDS via async b128/b32 copies -------------
  const float* img_base[2] = {pred + (size_t)b * 3 * HWC, gt + (size_t)b * 3 * HWC};
  for (int s = tid; s < TOTAL_SLOTS; s += NTHREADS) {
    const int p   = s / SLOTS_PER_PLANE;
    const int rem = s - p * SLOTS_PER_PLANE;
    const int r   = rem / SLOTS_PER_ROW;
    const int k   = rem - r * SLOTS_PER_ROW;
    const int gy  = ty0 + r - 1;
    if (gy < 0 || gy >= IMG_H) continue;            // stays zero (border only)
    const float* rowp = img_base[p >= 3] + (size_t)(p % 3) * HWC + (size_t)gy * IMG_W;
    float* ldsrow = &sm[p * PLANE + r * HSTR];
    if (k == 0) {                                   // left halo column
      if (tx0 > 0) async_copy_b32(rowp + (tx0 - 1), ldsrow + LOFF);
    } else if (k == SLOTS_PER_ROW - 1) {            // right halo column
      if (tx0 + TILE_W < IMG_W) async_copy_b32(rowp + tx0 + TILE_W, ldsrow + LOFF + 1 + TILE_W);
    } else {                                        // 16B-aligned interior chunk
      const int j = (k - 1) * 4;
      async_copy_b128(rowp + tx0 + j, ldsrow + LOFF + 1 + j);
    }
  }
  wait_async_zero();      // this wave's async LDS writes complete
  __syncthreads();        // all waves' writes visible to everyone

  // ---- fused: gradients -> cross product -> normalize -> masked L1 --------
  float loss_acc = 0.0f, mask_acc = 0.0f;
  for (int i = tid; i < TILE_H * TILE_W; i += NTHREADS) {
    const int py = i / TILE_W;
    const int px = i - py * TILE_W;
    const int r = py + 1;
    const int c = px + LOFF;      // LDS col of (px-1) halo element

    float nrm[2][3];
    #pragma unroll
    for (int img = 0; img < 2; ++img) {
      float gxv[3], gyv[3];
      #pragma unroll
      for (int ch = 0; ch < 3; ++ch) {
        const float* sp = &sm[(img * 3 + ch) * PLANE];
        const float a00 = sp[(r - 1) * HSTR + c    ];
        const float a01 = sp[(r - 1) * HSTR + c + 1];
        const float a02 = sp[(r - 1) * HSTR + c + 2];
        const float a10 = sp[(r    ) * HSTR + c    ];
        const float a12 = sp[(r    ) * HSTR + c + 2];
        const float a20 = sp[(r + 1) * HSTR + c    ];
        const float a21 = sp[(r + 1) * HSTR + c + 1];
        const float a22 = sp[(r + 1) * HSTR + c + 2];
        // SIZE=3 "Scharr" from reference degenerates to these separable sums
        gxv[ch] = 3.0f * ((a02 + a12 + a22) - (a00 + a10 + a20));
        gyv[ch] = 3.0f * ((a20 + a21 + a22) - (a00 + a01 + a02));
      }
      const float n0 = gxv[1] * gyv[2] - gxv[2] * gyv[1];
      const float n1 = gxv[2] * gyv[0] - gyv[2] * gxv[0];
      const float n2 = gxv[0] * gyv[1] - gxv[1] * gyv[0];
      const float inv = 1.0f / (sqrtf(n0 * n0 + n1 * n1 + n2 * n2) + 1e-10f);
      nrm[img][0] = n0 * inv; nrm[img][1] = n1 * inv; nrm[img][2] = n2 * inv;
    }
    const float m = mask[(size_t)b * HWC + (size_t)(ty0 + py) * IMG_W + (tx0 + px)];
    loss_acc += m * (fabsf(nrm[0][0] - nrm[1][0]) +
                     fabsf(nrm[0][1] - nrm[1][1]) +
                     fabsf(nrm[0][2] - nrm[1][2]));
    mask_acc += m;
  }

  // ---- deterministic block reduction --------------------------------------
  red[tid] = loss_acc;
  red[NTHREADS + tid] = mask_acc;
  __syncthreads();
  #pragma unroll
  for (int s = NTHREADS / 2; s > 0; s >>= 1) {
    if (tid < s) {
      red[tid]            += red[tid + s];
      red[NTHREADS + tid] += red[NTHREADS + tid + s];
    }
    __syncthreads();
  }
  if (tid == 0) {
    const int bid = blockIdx.x + gridDim.x * (blockIdx.y + gridDim.y * blockIdx.z);
    partial[bid]           = red[0];
    partial[nblocks + bid] = red[NTHREADS];
  }
}

__global__ __launch_bounds__(NTHREADS)
void reduce_final(const float* __restrict__ partial, int nblocks,
                  float* __restrict__ out) {
  __shared__ float red[2 * NTHREADS];
  float l = 0.0f, m = 0.0f;
  for (int i = threadIdx.x; i < nblocks; i += NTHREADS) {
    l += partial[i];
    m += partial[nblocks + i];
  }
  red[threadIdx.x] = l;
  red[NTHREADS + threadIdx.x] = m;
  __syncthreads();
  #pragma unroll
  for (int s = NTHREADS / 2; s > 0; s >>= 1) {
    if (threadIdx.x < s) {
      red[threadIdx.x]            += red[threadIdx.x + s];
      red[NTHREADS + threadIdx.x] += red[NTHREADS + threadIdx.x + s];
    }
    __syncthreads();
  }
  if (threadIdx.x == 0) out[0] = red[0] / red[NTHREADS];
}

extern "C" void kernel_launch(void* const* d_in, const int* in_sizes, int n_in,
                              void* d_out, int out_size, void* d_ws, size_t ws_size,
                              hipStream_t stream) {
  const float* pred = (const float*)d_in[0];
  const float* gt   = (const float*)d_in[1];
  const float* mask = (const float*)d_in[2];
  float* out = (float*)d_out;
  float* ws  = (float*)d_ws;   // needs 2 * nblocks floats (~60 KB)

  dim3 grid(IMG_W / TILE_W, IMG_H / TILE_H, IMG_B);   // 20 x 24 x 16 = 7680
  const int nblocks = grid.x * grid.y * grid.z;

  normal_loss_tiles<<<grid, NTHREADS, 0, stream>>>(pred, gt, mask, ws, nblocks);
  reduce_final<<<1, NTHREADS, 0, stream>>>(ws, nblocks, out);
}